// regresor_GIN_45088566673984
// MI455X (gfx1250) — compile-verified
//
#include <hip/hip_runtime.h>
#include <hip/hip_bf16.h>

typedef float v2f __attribute__((ext_vector_type(2)));
typedef float v8f __attribute__((ext_vector_type(8)));

#define SLOPE 0.01f

// lrelu(v) == max(v, SLOPE*v) for 0 < SLOPE < 1  ->  v_mul + v_max_num_f32,
// no v_cmp/v_cndmask chain.
__device__ __forceinline__ float lrelu(float v) { return fmaxf(v, SLOPE * v); }

// ---------------------------------------------------------------------------
// agg = src  (vectorized float4 copy; initializes the self-term of GIN agg)
// ---------------------------------------------------------------------------
__global__ void copy_f4_kernel(const float4* __restrict__ in, float4* __restrict__ out,
                               unsigned n4) {
    unsigned tid = blockIdx.x * blockDim.x + threadIdx.x;
    if (tid < n4) out[tid] = in[tid];
}

// ---------------------------------------------------------------------------
// agg[dst[e]][f] += feat[src[e]][f]   (COO scatter, lane-per-feature)
// F is a power of two (16 or 32) -> each edge touches one contiguous 64/128B
// segment; feature arrays are L2-resident (192MB L2), edge indices are
// streamed with non-temporal loads so they don't evict the features.
// ---------------------------------------------------------------------------
template <int LF>
__global__ void scatter_add_kernel(const float* __restrict__ feat,
                                   const int* __restrict__ eidx,   // [2][E]
                                   float* __restrict__ agg,
                                   unsigned long long nEdges) {
    unsigned long long tid = (unsigned long long)blockIdx.x * blockDim.x + threadIdx.x;
    unsigned long long total = nEdges << LF;
    if (tid >= total) return;
    unsigned long long e = tid >> LF;
    unsigned f = (unsigned)tid & ((1u << LF) - 1u);
    int s = __builtin_nontemporal_load(eidx + e);            // src row
    int d = __builtin_nontemporal_load(eidx + nEdges + e);   // dst row
    float v = feat[((size_t)s << LF) + f];
    unsafeAtomicAdd(&agg[((size_t)d << LF) + f], v);
}

// ---------------------------------------------------------------------------
// out[N x 32] = act( in[N x FIN] @ W[FIN x 32] + b )  via V_WMMA_F32_16X16X4_F32
// One wave handles a 16-node row tile. 8 waves / block. Weights in LDS.
// FIN / ACT are compile-time so afrag[] stays in static registers
// (no v_movrels / m0 indexing) and all loops fully unroll.
//
// f32 WMMA 16x16x4 fragment layouts (wave32):
//   A (16x4):  lanes 0-15 hold rows M=lane, {K=0,K=1} in VGPR0/1;
//              lanes 16-31 hold {K=2,K=3}.
//   B (4x16):  lanes 0-15 hold cols N=lane, {K=0,K=1}; lanes 16-31 {K=2,K=3}.
//   C/D (16x16): lanes 0-15: VGPR r -> M=r; lanes 16-31: VGPR r -> M=8+r; N=lane%16.
// ---------------------------------------------------------------------------
template <int FIN, bool ACT>
__global__ void gin_linear_kernel(const float* __restrict__ in,
                                  const float* __restrict__ W,   // [FIN][32] row-major
                                  const float* __restrict__ bias,// [32]
                                  float* __restrict__ out,       // [N][32]
                                  int n_nodes) {
    constexpr int KFRAGS = FIN / 4;        // 4 (FIN=16) or 8 (FIN=32)
    __shared__ float lds_w[FIN * 32];
    __shared__ float lds_b[32];

    const int tid = threadIdx.x;
    #pragma unroll
    for (int i = 0; i < (FIN * 32) / 256; ++i) lds_w[i * 256 + tid] = W[i * 256 + tid];
    if (tid < 32) lds_b[tid] = bias[tid];
    __syncthreads();

    const int wave = tid >> 5;
    const int lane = tid & 31;
    const int half = lane >> 4;    // 0: K={0,1}, 1: K={2,3} (A); same for B
    const int l    = lane & 15;    // row index (A) / col index (B,C,D)

    const int tile = blockIdx.x * (blockDim.x >> 5) + wave;
    const int row0 = tile * 16;
    if (row0 >= n_nodes) return;

    // Load A fragments once; reuse across both N-tiles.
    v2f afrag[KFRAGS];
    const float* rowp = in + (size_t)(row0 + l) * FIN + 2 * half;
    #pragma unroll
    for (int kf = 0; kf < KFRAGS; ++kf) {
        afrag[kf].x = rowp[4 * kf + 0];
        afrag[kf].y = rowp[4 * kf + 1];
    }

    #pragma unroll
    for (int nt = 0; nt < 2; ++nt) {
        const int col = nt * 16 + l;
        const float bv = lds_b[col];
        v8f acc;
        #pragma unroll
        for (int r = 0; r < 8; ++r) acc[r] = bv;

        #pragma unroll
        for (int kf = 0; kf < KFRAGS; ++kf) {
            const int k0 = 4 * kf + 2 * half;
            v2f bfrag;
            bfrag.x = lds_w[(k0 + 0) * 32 + col];
            bfrag.y = lds_w[(k0 + 1) * 32 + col];
            acc = __builtin_amdgcn_wmma_f32_16x16x4_f32(
                false, afrag[kf], false, bfrag, (short)0, acc, false, false);
        }

        if (ACT) {
            #pragma unroll
            for (int r = 0; r < 8; ++r) acc[r] = lrelu(acc[r]);
        }

        #pragma unroll
        for (int r = 0; r < 8; ++r) {
            out[(size_t)(row0 + half * 8 + r) * 32 + col] = acc[r];
        }
    }
}

// ---------------------------------------------------------------------------
// y[g] = fc_b + sum_f lrelu(h3[g*nPer + 7][f]) * fc_w[f]
// ---------------------------------------------------------------------------
__global__ void head_kernel(const float* __restrict__ h3,
                            const float* __restrict__ fc_w,
                            const float* __restrict__ fc_b,
                            float* __restrict__ y,
                            int n_graphs, int n_per_graph) {
    int g = blockIdx.x * blockDim.x + threadIdx.x;
    if (g >= n_graphs) return;
    const float* row = h3 + (size_t)(g * n_per_graph + 7) * 32;
    float s = fc_b[0];
    #pragma unroll
    for (int f = 0; f < 32; ++f) s += lrelu(row[f]) * fc_w[f];
    y[g] = s;
}

// ---------------------------------------------------------------------------
extern "C" void kernel_launch(void* const* d_in, const int* in_sizes, int n_in,
                              void* d_out, int out_size, void* d_ws, size_t ws_size,
                              hipStream_t stream) {
    const float* x    = (const float*)d_in[0];
    const int*   eidx = (const int*)  d_in[1];
    // d_in[2] = batch_size scalar (device); use out_size instead on host.
    const float* w1_1 = (const float*)d_in[3];
    const float* b1_1 = (const float*)d_in[4];
    const float* w2_1 = (const float*)d_in[5];
    const float* b2_1 = (const float*)d_in[6];
    const float* w1_2 = (const float*)d_in[7];
    const float* b1_2 = (const float*)d_in[8];
    const float* w2_2 = (const float*)d_in[9];
    const float* b2_2 = (const float*)d_in[10];
    const float* w1_3 = (const float*)d_in[11];
    const float* b1_3 = (const float*)d_in[12];
    const float* w2_3 = (const float*)d_in[13];
    const float* b2_3 = (const float*)d_in[14];
    const float* fc_w = (const float*)d_in[15];
    const float* fc_b = (const float*)d_in[16];

    const int N = in_sizes[0] / 16;                       // 512000 nodes
    const unsigned long long E = (unsigned long long)in_sizes[1] / 2; // 8M edges
    const int B = out_size;                               // 256 graphs
    const int nPer = N / B;                               // 2000 nodes/graph

    float* bufA = (float*)d_ws;                          // agg  : N x 32 f32
    float* bufB = bufA + (size_t)N * 32;                 // z    : N x 32 f32
    float* bufC = bufB + (size_t)N * 32;                 // h    : N x 32 f32

    const int TPB = 256;
    const int linBlocks = (N + 16 * 8 - 1) / (16 * 8);   // 1 wave per 16-node tile

    // ---- Layer 1 (fin=16) ----
    {
        unsigned n4 = (unsigned)N * 16 / 4;
        copy_f4_kernel<<<(n4 + TPB - 1) / TPB, TPB, 0, stream>>>(
            (const float4*)x, (float4*)bufA, n4);
        unsigned long long tot = E << 4;
        scatter_add_kernel<4><<<(unsigned)((tot + TPB - 1) / TPB), TPB, 0, stream>>>(
            x, eidx, bufA, E);
        gin_linear_kernel<16, true><<<linBlocks, TPB, 0, stream>>>(bufA, w1_1, b1_1, bufB, N);
        gin_linear_kernel<32, true><<<linBlocks, TPB, 0, stream>>>(bufB, w2_1, b2_1, bufC, N);
    }
    // ---- Layer 2 (fin=32) ----
    {
        unsigned n4 = (unsigned)N * 32 / 4;
        copy_f4_kernel<<<(n4 + TPB - 1) / TPB, TPB, 0, stream>>>(
            (const float4*)bufC, (float4*)bufA, n4);
        unsigned long long tot = E << 5;
        scatter_add_kernel<5><<<(unsigned)((tot + TPB - 1) / TPB), TPB, 0, stream>>>(
            bufC, eidx, bufA, E);
        gin_linear_kernel<32, true><<<linBlocks, TPB, 0, stream>>>(bufA, w1_2, b1_2, bufB, N);
        gin_linear_kernel<32, true><<<linBlocks, TPB, 0, stream>>>(bufB, w2_2, b2_2, bufC, N);
    }
    // ---- Layer 3 (fin=32, no trailing act) ----
    {
        unsigned n4 = (unsigned)N * 32 / 4;
        copy_f4_kernel<<<(n4 + TPB - 1) / TPB, TPB, 0, stream>>>(
            (const float4*)bufC, (float4*)bufA, n4);
        unsigned long long tot = E << 5;
        scatter_add_kernel<5><<<(unsigned)((tot + TPB - 1) / TPB), TPB, 0, stream>>>(
            bufC, eidx, bufA, E);
        gin_linear_kernel<32, true><<<linBlocks, TPB, 0, stream>>>(bufA, w1_3, b1_3, bufB, N);
        gin_linear_kernel<32, false><<<linBlocks, TPB, 0, stream>>>(bufB, w2_3, b2_3, bufC, N);
    }
    // ---- Head ----
    head_kernel<<<(B + TPB - 1) / TPB, TPB, 0, stream>>>(bufC, fc_w, fc_b, (float*)d_out,
                                                         B, nPer);
}